// SnailBlock_88244398064272
// MI455X (gfx1250) — compile-verified
//
#include <hip/hip_runtime.h>
#include <hip/hip_bf16.h>

// SnailBlock forward for MI455X (gfx1250), fp32 WMMA (V_WMMA_F32_16X16X4_F32).
// All convs cast as GEMMs on the wave32 matrix pipe; flash attention with
// online softmax; LDS P-tile relayout with explicit s_wait_dscnt.

typedef __attribute__((ext_vector_type(2))) float v2f;
typedef __attribute__((ext_vector_type(8))) float v8f;

#define NB   32
#define HIDC 128
#define HW   1024

__device__ __forceinline__ v8f wmma4(v2f a, v2f b, v8f c) {
  // D = A(16x4 f32) * B(4x16 f32) + C(16x16 f32)
  return __builtin_amdgcn_wmma_f32_16x16x4_f32(false, a, false, b, (short)0, c,
                                               false, false);
}

__device__ __forceinline__ float eluf(float x) {
  return x > 0.f ? x : (expf(x) - 1.f);
}

// ---------------- elementwise ----------------
__global__ void k_elu(const float* __restrict__ X, float* __restrict__ Y, int n) {
  int i = blockIdx.x * 256 + threadIdx.x;
  if (i < n) Y[i] = eluf(X[i]);
}

__global__ void k_add_elu(const float* __restrict__ A, const float* __restrict__ B,
                          float* __restrict__ Y, int n) {
  int i = blockIdx.x * 256 + threadIdx.x;
  if (i < n) Y[i] = eluf(A[i] + B[i]);
}

// out += t1 * sigmoid(t2); T2 is [b][256][1024], out is [b][128][1024]
__global__ void k_gate(float* __restrict__ out, const float* __restrict__ T2) {
  int i = blockIdx.x * 256 + threadIdx.x;  // over NB*128*1024
  int b = i >> 17;
  int rem = i & 0x1FFFF;
  float t1 = T2[((size_t)b << 18) + rem];
  float t2 = T2[((size_t)b << 18) + (1 << 17) + rem];
  out[i] += t1 * (1.f / (1.f + expf(-t2)));
}

// ---------------- 2x2 conv as 4 shifted GEMMs ----------------
// X: [b][128][1024] (pre-ELU'd), W: [O][128][2][2], Y: [b][O][1024]
// grid: (NB, HW/64, O/128), block 256 (8 waves); wave = 16 O-rows x 64 px.
__global__ void __launch_bounds__(256)
k_conv2x2(const float* __restrict__ X, const float* __restrict__ W,
          const float* __restrict__ bias, float* __restrict__ Y, int O) {
  const int lane = threadIdx.x & 31;
  const int wave = threadIdx.x >> 5;
  const int half = lane >> 4;
  const int l15  = lane & 15;
  const int b  = blockIdx.x;
  const int p0 = blockIdx.y * 64;
  const int o0 = (blockIdx.z * 8 + wave) * 16;
  if (o0 >= O) return;

  const float* Xb = X + (size_t)b * HIDC * HW;

  int   sidx[4][4];
  float smsk[4][4];
#pragma unroll
  for (int t = 0; t < 4; ++t) {
    int p = p0 + t * 16 + l15;
    int y = p >> 5, x = p & 31;
#pragma unroll
    for (int tap = 0; tap < 4; ++tap) {
      int ys = y + (tap >> 1) - 1;
      int xs = x + (tap & 1) - 1;
      bool ok = (ys >= 0) && (xs >= 0);
      sidx[t][tap] = ok ? (ys * 32 + xs) : 0;
      smsk[t][tap] = ok ? 1.f : 0.f;
    }
  }

  v8f acc[4];
#pragma unroll
  for (int t = 0; t < 4; ++t) acc[t] = (v8f){0, 0, 0, 0, 0, 0, 0, 0};

  const int orow = o0 + l15;
#pragma unroll
  for (int tap = 0; tap < 4; ++tap) {
    const float* wt = W + (size_t)orow * 512 + tap;  // o*512 + c*4 + tap
    for (int c0 = 0; c0 < HIDC; c0 += 4) {
      int ca = c0 + half * 2;
      v2f a;
      a.x = wt[(size_t)ca * 4];
      a.y = wt[(size_t)(ca + 1) * 4];
      const float* x0 = Xb + (size_t)ca * HW;
      const float* x1 = x0 + HW;
#pragma unroll
      for (int t = 0; t < 4; ++t) {
        v2f bb;
        bb.x = x0[sidx[t][tap]] * smsk[t][tap];
        bb.y = x1[sidx[t][tap]] * smsk[t][tap];
        acc[t] = wmma4(a, bb, acc[t]);
      }
    }
  }

#pragma unroll
  for (int t = 0; t < 4; ++t) {
#pragma unroll
    for (int r = 0; r < 8; ++r) {
      int o = o0 + r + half * 8;
      float v = acc[t][r] + bias[o];
      Y[((size_t)b * O + o) * HW + p0 + t * 16 + l15] = v;
    }
  }
}

// ---------------- generic 1x1 conv GEMM (Cin = 128) ----------------
// Y[b][o][p] = (elu?)( W[o][:]·X[b][:][p] + bias[o] ) (+ R[b][o][p])
__global__ void __launch_bounds__(256)
k_gemm1x1(const float* __restrict__ X, const float* __restrict__ W,
          const float* __restrict__ bias, float* __restrict__ Y,
          const float* __restrict__ R, int O, int outElu) {
  const int lane = threadIdx.x & 31;
  const int wave = threadIdx.x >> 5;
  const int half = lane >> 4;
  const int l15  = lane & 15;
  const int b  = blockIdx.x;
  const int p0 = blockIdx.y * 64;
  const int o0 = (blockIdx.z * 8 + wave) * 16;
  if (o0 >= O) return;

  const float* Xb   = X + (size_t)b * HIDC * HW;
  const float* wrow = W + (size_t)(o0 + l15) * HIDC;

  v8f acc[4];
#pragma unroll
  for (int t = 0; t < 4; ++t) acc[t] = (v8f){0, 0, 0, 0, 0, 0, 0, 0};

  for (int c0 = 0; c0 < HIDC; c0 += 4) {
    int ca = c0 + half * 2;
    v2f a;
    a.x = wrow[ca];
    a.y = wrow[ca + 1];
    const float* x0 = Xb + (size_t)ca * HW + p0 + l15;
#pragma unroll
    for (int t = 0; t < 4; ++t) {
      v2f bb;
      bb.x = x0[t * 16];
      bb.y = x0[HW + t * 16];
      acc[t] = wmma4(a, bb, acc[t]);
    }
  }

#pragma unroll
  for (int t = 0; t < 4; ++t) {
#pragma unroll
    for (int r = 0; r < 8; ++r) {
      int o = o0 + r + half * 8;
      size_t idx = ((size_t)b * O + o) * HW + p0 + t * 16 + l15;
      float v = acc[t][r] + bias[o];
      if (outElu) v = eluf(v);
      if (R) v += R[idx];
      Y[idx] = v;
    }
  }
}

// ---------------- qkv 1x1 conv with positional-encoding epilogue ----------------
// W: [160][130] (cols 128,129 are py,px weights); Y: [b][160][1024]
__global__ void __launch_bounds__(256)
k_qkv(const float* __restrict__ X, const float* __restrict__ W,
      const float* __restrict__ bias, float* __restrict__ Y) {
  const int lane = threadIdx.x & 31;
  const int wave = threadIdx.x >> 5;
  const int half = lane >> 4;
  const int l15  = lane & 15;
  const int b  = blockIdx.x;
  const int p0 = blockIdx.y * 64;
  const int o0 = (blockIdx.z * 8 + wave) * 16;
  if (o0 >= 160) return;

  const float* Xb   = X + (size_t)b * HIDC * HW;
  const float* wrow = W + (size_t)(o0 + l15) * 130;

  v8f acc[4];
#pragma unroll
  for (int t = 0; t < 4; ++t) acc[t] = (v8f){0, 0, 0, 0, 0, 0, 0, 0};

  for (int c0 = 0; c0 < HIDC; c0 += 4) {
    int ca = c0 + half * 2;
    v2f a;
    a.x = wrow[ca];
    a.y = wrow[ca + 1];
    const float* x0 = Xb + (size_t)ca * HW + p0 + l15;
#pragma unroll
    for (int t = 0; t < 4; ++t) {
      v2f bb;
      bb.x = x0[t * 16];
      bb.y = x0[HW + t * 16];
      acc[t] = wmma4(a, bb, acc[t]);
    }
  }

#pragma unroll
  for (int t = 0; t < 4; ++t) {
    int p = p0 + t * 16 + l15;
    float py = (float)(p >> 5) * (1.f / 32.f) - 0.5f;
    float px = (float)(p & 31) * (1.f / 32.f) - 0.5f;
#pragma unroll
    for (int r = 0; r < 8; ++r) {
      int o = o0 + r + half * 8;
      float v = acc[t][r] + W[(size_t)o * 130 + 128] * py +
                W[(size_t)o * 130 + 129] * px + bias[o];
      Y[((size_t)b * 160 + o) * HW + p] = v;
    }
  }
}

// ---------------- causal flash attention (QK=16, V=128, HW=1024) ----------------
// QKV: [b][160][1024] (ch 0-15 Q, 16-31 K, 32-159 V). Ob: [b][128][1024].
// grid: (64 qtiles, NB), block 32 (one wave).
__global__ void __launch_bounds__(32)
k_attn(const float* __restrict__ QKV, float* __restrict__ Ob) {
  __shared__ float pl[16 * 17];
  const int lane = threadIdx.x;
  const int half = lane >> 4;
  const int l15  = lane & 15;
  const int qt = blockIdx.x;
  const int b  = blockIdx.y;
  const int q0 = qt * 16;
  const float* Qb = QKV + (size_t)b * 160 * HW;

  // Q A-fragments (A[m][k] = Qchan[k][q0+m]), 4 chunks of K=4 over d=16
  v2f qa[4];
#pragma unroll
  for (int kc = 0; kc < 4; ++kc) {
    qa[kc].x = Qb[(size_t)(kc * 4 + half * 2) * HW + q0 + l15];
    qa[kc].y = Qb[(size_t)(kc * 4 + half * 2 + 1) * HW + q0 + l15];
  }

  v8f acc[8];
#pragma unroll
  for (int vt = 0; vt < 8; ++vt) acc[vt] = (v8f){0, 0, 0, 0, 0, 0, 0, 0};
  float mrun[8], lrun[8];
#pragma unroll
  for (int r = 0; r < 8; ++r) { mrun[r] = -3.0e38f; lrun[r] = 0.f; }

  for (int kt = 0; kt <= qt; ++kt) {
    int k0 = kt * 16;
    // scores S = Q·K^T
    v8f s = (v8f){0, 0, 0, 0, 0, 0, 0, 0};
#pragma unroll
    for (int kc = 0; kc < 4; ++kc) {
      v2f bb;
      bb.x = Qb[(size_t)(16 + kc * 4 + half * 2) * HW + k0 + l15];
      bb.y = Qb[(size_t)(16 + kc * 4 + half * 2 + 1) * HW + k0 + l15];
      s = wmma4(qa[kc], bb, s);
    }
    // online softmax per row (row m = r + half*8, cols across 16 lanes of half)
#pragma unroll
    for (int r = 0; r < 8; ++r) {
      int qp = q0 + r + half * 8;
      int kp = k0 + l15;
      float val = (kp <= qp) ? s[r] * 0.25f : -3.0e38f;
      float rm = val;
      rm = fmaxf(rm, __shfl_xor(rm, 1));
      rm = fmaxf(rm, __shfl_xor(rm, 2));
      rm = fmaxf(rm, __shfl_xor(rm, 4));
      rm = fmaxf(rm, __shfl_xor(rm, 8));
      float nm = fmaxf(mrun[r], rm);
      float alpha = expf(mrun[r] - nm);
      float p = expf(val - nm);
      float rs = p;
      rs += __shfl_xor(rs, 1);
      rs += __shfl_xor(rs, 2);
      rs += __shfl_xor(rs, 4);
      rs += __shfl_xor(rs, 8);
      lrun[r] = lrun[r] * alpha + rs;
      mrun[r] = nm;
#pragma unroll
      for (int vt = 0; vt < 8; ++vt) acc[vt][r] *= alpha;
      pl[(r + half * 8) * 17 + l15] = p;  // C/D layout -> LDS
    }
    asm volatile("s_wait_dscnt 0" ::: "memory");  // cross-lane LDS visibility
    // O += P·V  (A = P relayout from LDS, B = V^T fragments)
#pragma unroll
    for (int kc = 0; kc < 4; ++kc) {
      v2f pa;
      pa.x = pl[l15 * 17 + kc * 4 + half * 2];
      pa.y = pl[l15 * 17 + kc * 4 + half * 2 + 1];
#pragma unroll
      for (int vt = 0; vt < 8; ++vt) {
        const float* vrow =
            Qb + (size_t)(32 + vt * 16 + l15) * HW + k0 + kc * 4 + half * 2;
        v2f bb;
        bb.x = vrow[0];
        bb.y = vrow[1];
        acc[vt] = wmma4(pa, bb, acc[vt]);
      }
    }
  }

  // normalize and store O^T as channel-major [b][128][1024]
#pragma unroll
  for (int r = 0; r < 8; ++r) {
    float inv = 1.f / lrun[r];
    int qp = q0 + r + half * 8;
#pragma unroll
    for (int vt = 0; vt < 8; ++vt) {
      Ob[((size_t)b * HIDC + vt * 16 + l15) * HW + qp] = acc[vt][r] * inv;
    }
  }
}

// ---------------- launcher ----------------
extern "C" void kernel_launch(void* const* d_in, const int* in_sizes, int n_in,
                              void* d_out, int out_size, void* d_ws, size_t ws_size,
                              hipStream_t stream) {
  const float* x    = (const float*)d_in[0];
  const float* rw1  = (const float*)d_in[1];
  const float* rb1  = (const float*)d_in[2];
  const float* rw2  = (const float*)d_in[3];
  const float* rb2  = (const float*)d_in[4];
  const float* qkvw = (const float*)d_in[5];
  const float* qkvb = (const float*)d_in[6];
  const float* apw  = (const float*)d_in[7];
  const float* apb  = (const float*)d_in[8];
  const float* ocw  = (const float*)d_in[9];
  const float* ocb  = (const float*)d_in[10];
  const float* oaw  = (const float*)d_in[11];
  const float* oab  = (const float*)d_in[12];
  const float* opw  = (const float*)d_in[13];
  const float* opb  = (const float*)d_in[14];
  float* Yout = (float*)d_out;

  const size_t N1 = (size_t)NB * HIDC * HW;  // 4,194,304 floats (16.8 MB)
  float* A  = (float*)d_ws;   // residual state / conv     (N1)
  float* Bf = A + N1;         // scratch                   (N1)
  float* Cf = Bf + N1;        // scratch                   (N1)
  float* Df = Cf + N1;        // wide scratch (T2 / qkv)   (2*N1)

  hipMemcpyAsync(A, x, N1 * sizeof(float), hipMemcpyDeviceToDevice, stream);

  dim3 blk(256);
  dim3 ew((unsigned)((N1 + 255) / 256));
  dim3 g128(NB, HW / 64, 1), g256(NB, HW / 64, 2);

  // ---- residual gated-conv stack ----
  for (int i = 0; i < 6; ++i) {
    const float* w1 = rw1 + (size_t)i * HIDC * HIDC * 4;
    const float* b1 = rb1 + (size_t)i * HIDC;
    const float* w2 = rw2 + (size_t)i * 2 * HIDC * HIDC * 4;
    const float* b2 = rb2 + (size_t)i * 2 * HIDC;
    k_elu<<<ew, blk, 0, stream>>>(A, Bf, (int)N1);
    k_conv2x2<<<g128, blk, 0, stream>>>(Bf, w1, b1, Cf, HIDC);
    k_elu<<<ew, blk, 0, stream>>>(Cf, Bf, (int)N1);
    k_conv2x2<<<g256, blk, 0, stream>>>(Bf, w2, b2, Df, 2 * HIDC);
    k_gate<<<ew, blk, 0, stream>>>(A, Df);
  }

  // ---- attention branch ----
  k_qkv<<<g256, blk, 0, stream>>>(A, qkvw, qkvb, Df);            // Df = qkv
  k_attn<<<dim3(HW / 16, NB), dim3(32), 0, stream>>>(Df, Cf);    // Cf = o
  k_gemm1x1<<<g128, blk, 0, stream>>>(Cf, apw, apb, Bf, nullptr, HIDC, 0);  // Bf = attn_out

  // ---- conv branch: c1 = elu(Wc·elu(conv)+b) ----
  k_elu<<<ew, blk, 0, stream>>>(A, Cf, (int)N1);
  k_gemm1x1<<<g128, blk, 0, stream>>>(Cf, ocw, ocb, Df, nullptr, HIDC, 1);  // Df = c1

  // ---- attn branch: a1 = elu(Wa·elu(attn_out)+b) ----
  k_elu<<<ew, blk, 0, stream>>>(Bf, Cf, (int)N1);
  k_gemm1x1<<<g128, blk, 0, stream>>>(Cf, oaw, oab, Bf, nullptr, HIDC, 1);  // Bf = a1

  // ---- combine: y = elu(Wp·elu(c1+a1)+b) + x ----
  k_add_elu<<<ew, blk, 0, stream>>>(Df, Bf, Cf, (int)N1);
  k_gemm1x1<<<g128, blk, 0, stream>>>(Cf, opw, opb, Yout, x, HIDC, 1);
}